// CP_LIF_16131897164004
// MI455X (gfx1250) — compile-verified
//
#include <hip/hip_runtime.h>
#include <cstdint>

// LIF spiking-neuron scan, MI455X (gfx1250).
// Bandwidth-bound streaming kernel: 420 MB traffic @ 23.3 TB/s -> ~18us floor.
// CDNA5 async global->LDS loads (ASYNCcnt) as an 8-deep prefetch pipeline,
// non-temporal b128 stores for the write-once output, float4 per thread.

#define T_FIXED   100
#define V_TH      1.0f
#define TAU_MIN   1e-3f
#define R_MIN     1e-6f

typedef float v4f __attribute__((ext_vector_type(4)));

__device__ __forceinline__ float sp_f(float x) {
    // numerically stable softplus: max(x,0) + log1p(exp(-|x|))
    return fmaxf(x, 0.0f) + log1pf(expf(-fabsf(x)));
}

__global__ __launch_bounds__(256) void lif_scan_kernel(
    const float* __restrict__ x,        // (T, B, N)
    const float* __restrict__ tau_raw,  // (N,)
    const float* __restrict__ r_raw,    // (N,)
    float* __restrict__ out,            // (T, B, N)
    int T, int BN, int N)
{
    constexpr int STAGES = 8;                 // async pipeline depth (~1.4us coverage)
    __shared__ v4f lds[STAGES][256];          // 32 KB per block

    const int tid = threadIdx.x;
    const int gid = blockIdx.x * 256 + tid;   // one float4 chunk of (B,N)
    const int nChunks = BN >> 2;
    if (gid >= nChunks) return;

    // ---- per-neuron parameters (computed once, amortized over T steps) ----
    const int rowChunks = N >> 2;
    const int n4 = ((rowChunks & (rowChunks - 1)) == 0)
                       ? (gid & (rowChunks - 1))      // power-of-two fast path
                       : (gid % rowChunks);
    v4f tau4 = *(const v4f*)(tau_raw + (size_t)n4 * 4);
    v4f r4   = *(const v4f*)(r_raw   + (size_t)n4 * 4);
    v4f al, om, rs;
#pragma unroll
    for (int j = 0; j < 4; ++j) {
        float tau = sp_f(tau4[j]) + TAU_MIN;
        al[j] = expf(-1.0f / tau);            // decay
        om[j] = 1.0f - al[j];
        rs[j] = sp_f(r4[j]) + R_MIN;          // soft-reset amount
    }

    // ---- addresses ----
    const uint32_t stepBytes = (uint32_t)BN * 4u;     // bytes per timestep slab
    uint32_t voff = (uint32_t)gid * 16u;              // byte offset into x for next prefetch
    // LDS byte offset of this lane's stage-0 slot (addrspace(3) low 32 bits)
    const uint32_t lane_lds = (uint32_t)(uintptr_t)(&lds[0][tid]);
    float* op = out + (size_t)gid * 4;                // output cursor

    // ---- prologue: fill the async pipeline ----
    const int P = (T < STAGES) ? T : STAGES;
    for (int i = 0; i < P; ++i) {
        uint32_t l = lane_lds + ((uint32_t)(i & (STAGES - 1)) << 12); // 4096 B / stage
        asm volatile(
            "global_load_async_to_lds_b128 %0, %1, %2 th:TH_LOAD_NT\n"
            :
            : "v"(l), "v"(voff), "s"(x)
            : "memory");
        voff += stepBytes;
    }

    v4f vst = {0.f, 0.f, 0.f, 0.f};   // membrane state
    int t = 0;
    const int mainT = (T > STAGES) ? (T - STAGES) : 0;

    // ---- steady state: consume stage t, refill with t+STAGES ----
#pragma unroll 2
    for (; t < mainT; ++t) {
        // STAGES loads outstanding; <=STAGES-1 left => the oldest (step t) landed.
        asm volatile("s_wait_asynccnt 0x7" ::: "memory");
        v4f x4 = lds[t & (STAGES - 1)][tid];

        v4f s4;
#pragma unroll
        for (int j = 0; j < 4; ++j) {
            float v = fmaf(al[j], vst[j], om[j] * x4[j]);
            float s = (v > V_TH) ? 1.0f : 0.0f;
            s4[j]  = s;
            vst[j] = fmaf(-s, rs[j], v);       // soft reset
        }
        __builtin_nontemporal_store(s4, (v4f*)op);   // write-once: th NT
        op += BN;

        // Reissue into the slot just consumed; s_wait_dscnt 0 closes the
        // cross-counter (DS read vs ASYNC LDS write) WAR hazard.
        uint32_t l = lane_lds + ((uint32_t)(t & (STAGES - 1)) << 12);
        asm volatile(
            "s_wait_dscnt 0x0\n\t"
            "global_load_async_to_lds_b128 %0, %1, %2 th:TH_LOAD_NT\n"
            :
            : "v"(l), "v"(voff), "s"(x)
            : "memory");
        voff += stepBytes;
    }

    // ---- drain tail (last STAGES steps): wait for everything outstanding ----
    for (; t < T; ++t) {
        asm volatile("s_wait_asynccnt 0x0" ::: "memory");
        v4f x4 = lds[t & (STAGES - 1)][tid];

        v4f s4;
#pragma unroll
        for (int j = 0; j < 4; ++j) {
            float v = fmaf(al[j], vst[j], om[j] * x4[j]);
            float s = (v > V_TH) ? 1.0f : 0.0f;
            s4[j]  = s;
            vst[j] = fmaf(-s, rs[j], v);
        }
        __builtin_nontemporal_store(s4, (v4f*)op);
        op += BN;
    }
}

extern "C" void kernel_launch(void* const* d_in, const int* in_sizes, int n_in,
                              void* d_out, int out_size, void* d_ws, size_t ws_size,
                              hipStream_t stream) {
    (void)n_in; (void)out_size; (void)d_ws; (void)ws_size;
    const float* x       = (const float*)d_in[0];
    const float* tau_raw = (const float*)d_in[1];
    const float* r_raw   = (const float*)d_in[2];
    float* out = (float*)d_out;

    const int N     = in_sizes[1];          // 4096
    const int total = in_sizes[0];          // T*B*N
    const int T     = T_FIXED;              // reference fixes T=100
    const int BN    = total / T;            // 524288

    const int chunks = BN >> 2;             // float4 per thread
    const int block  = 256;
    const int grid   = (chunks + block - 1) / block;   // 512

    lif_scan_kernel<<<grid, block, 0, stream>>>(x, tau_raw, r_raw, out, T, BN, N);
}